// RelationMessagePassingBase_10170482557014
// MI455X (gfx1250) — compile-verified
//
#include <hip/hip_runtime.h>
#include <hip/hip_bf16.h>
#include <math.h>

typedef float v2f __attribute__((ext_vector_type(2)));
typedef float v8f __attribute__((ext_vector_type(8)));

__device__ __forceinline__ float mish_f(float x) {
    float sp = (x > 20.0f) ? x : log1pf(expf(x));   // softplus
    return x * tanhf(sp);
}

// D = ARITY*64.  One block = 32 atom-rows.  128 threads = 4 wave32s.
// Each wave owns n-tiles and computes TWO 16x16 output tiles per n-tile
// (rows 0-15 and 16-31) so every B-fragment load feeds two WMMAs.
template <int ARITY>
__global__ __launch_bounds__(128) void rel_msg_kernel(
    const float* __restrict__ node_emb,   // [N, 64]
    const int*   __restrict__ idx,        // [num_atoms * ARITY]
    const float* __restrict__ w_in,       // [D, D] row-major
    const float* __restrict__ b_in,       // [D]
    const float* __restrict__ w_out,      // [D, D]
    const float* __restrict__ b_out,      // [D]
    float*       __restrict__ out,        // [num_atoms, D]
    int num_atoms)
{
    constexpr int D  = ARITY * 64;
    constexpr int S  = D + 4;        // LDS row stride (floats): pad -> conflict-free
    constexpr int NT = D / 16;       // 16-wide column tiles

    __shared__ float Xs[32 * S];
    __shared__ float Hs[32 * S];

    const int tid   = threadIdx.x;
    const int wave  = tid >> 5;
    const int lane  = tid & 31;
    const int lo    = lane & 15;     // N column within tile / A-matrix row
    const int hi    = lane >> 4;     // K sub-pair (A/B) or M+8 (C/D)
    const int atom0 = blockIdx.x * 32;

    // ---- async gather straight into LDS: X[row, a*64+j] = node_emb[idx[...]][j]
    constexpr int RV = D / 4;                  // float4s per row
    for (int t = tid; t < 32 * RV; t += 128) {
        int row = t / RV;
        int col = (t - row * RV) * 4;
        int a   = col >> 6;
        int j   = col & 63;
        int ar  = atom0 + row;
        if (ar >= num_atoms) ar = num_atoms - 1;
        int node = idx[ar * ARITY + a];
        unsigned lds_off = (unsigned)(uintptr_t)(&Xs[row * S + col]);
        unsigned long long ga =
            (unsigned long long)(uintptr_t)(node_emb + (size_t)node * 64 + j);
        asm volatile("global_load_async_to_lds_b128 %0, %1, off"
                     :: "v"(lds_off), "v"(ga) : "memory");
    }
    asm volatile("s_wait_asynccnt 0" ::: "memory");
    __syncthreads();

    // ---- GEMM1: H = mish(X @ Win^T + b_in)
    for (int nt = wave; nt < NT; nt += 4) {
        const int n0 = nt * 16;
        const float bin = b_in[n0 + lo];
        v8f acc0 = {bin, bin, bin, bin, bin, bin, bin, bin};
        v8f acc1 = acc0;
        const float* wrow  = w_in + (size_t)(n0 + lo) * D + 2 * hi; // Win^T[k,n]=Win[n,k]
        const float* xrow0 = Xs + lo * S + 2 * hi;                  // rows 0-15
        const float* xrow1 = Xs + (16 + lo) * S + 2 * hi;           // rows 16-31
        for (int k = 0; k < D; k += 4) {
            v2f bv = *(const v2f*)(wrow + k);
            v2f a0 = *(const v2f*)(xrow0 + k);
            v2f a1 = *(const v2f*)(xrow1 + k);
            acc0 = __builtin_amdgcn_wmma_f32_16x16x4_f32(
                false, a0, false, bv, (short)0, acc0, false, false);
            acc1 = __builtin_amdgcn_wmma_f32_16x16x4_f32(
                false, a1, false, bv, (short)0, acc1, false, false);
        }
#pragma unroll
        for (int r = 0; r < 8; ++r) {
            int m = r + 8 * hi;
            Hs[m * S + n0 + lo]        = mish_f(acc0[r]);
            Hs[(16 + m) * S + n0 + lo] = mish_f(acc1[r]);
        }
    }
    __syncthreads();   // GEMM2's K-loop spans every wave's H columns

    // ---- GEMM2: Y = X + H @ Wout^T + b_out
    for (int nt = wave; nt < NT; nt += 4) {
        const int n0 = nt * 16;
        const float bo = b_out[n0 + lo];
        v8f acc0, acc1;
#pragma unroll
        for (int r = 0; r < 8; ++r) {
            int m = r + 8 * hi;
            acc0[r] = Xs[m * S + n0 + lo] + bo;          // residual + bias
            acc1[r] = Xs[(16 + m) * S + n0 + lo] + bo;
        }
        const float* wrow  = w_out + (size_t)(n0 + lo) * D + 2 * hi;
        const float* hrow0 = Hs + lo * S + 2 * hi;
        const float* hrow1 = Hs + (16 + lo) * S + 2 * hi;
        for (int k = 0; k < D; k += 4) {
            v2f bv = *(const v2f*)(wrow + k);
            v2f a0 = *(const v2f*)(hrow0 + k);
            v2f a1 = *(const v2f*)(hrow1 + k);
            acc0 = __builtin_amdgcn_wmma_f32_16x16x4_f32(
                false, a0, false, bv, (short)0, acc0, false, false);
            acc1 = __builtin_amdgcn_wmma_f32_16x16x4_f32(
                false, a1, false, bv, (short)0, acc1, false, false);
        }
#pragma unroll
        for (int r = 0; r < 8; ++r) {
            int m0 = atom0 + r + 8 * hi;
            int m1 = m0 + 16;
            if (m0 < num_atoms) out[(size_t)m0 * D + n0 + lo] = acc0[r];
            if (m1 < num_atoms) out[(size_t)m1 * D + n0 + lo] = acc1[r];
        }
    }
}

__global__ void copy_idx32(const int* __restrict__ src, int n, int* __restrict__ dst) {
    int i = blockIdx.x * 256 + threadIdx.x;
    if (i < n) dst[i] = src[i];
}
__global__ void copy_idx64(const int* __restrict__ src, int n, long long* __restrict__ dst) {
    int i = blockIdx.x * 256 + threadIdx.x;
    if (i < n) dst[i] = (long long)src[i];
}

extern "C" void kernel_launch(void* const* d_in, const int* in_sizes, int n_in,
                              void* d_out, int out_size, void* d_ws, size_t ws_size,
                              hipStream_t stream) {
    (void)n_in; (void)d_ws; (void)ws_size;

    const float* node = (const float*)d_in[0];
    const int* atoms[4] = {(const int*)d_in[1], (const int*)d_in[2],
                           (const int*)d_in[3], (const int*)d_in[4]};
    const float* W[4][4];
    for (int r = 0; r < 4; ++r)
        for (int p = 0; p < 4; ++p)
            W[r][p] = (const float*)d_in[5 + 4 * r + p];

    const int arity[4] = {1, 2, 2, 3};
    float* out = (float*)d_out;

    size_t msg_base = 0;   // float offset into d_out
    for (int r = 0; r < 4; ++r) {
        int flat  = in_sizes[1 + r];          // num_atoms * arity
        int natom = flat / arity[r];
        dim3 grid((natom + 31) / 32), block(128);
        switch (arity[r]) {
            case 1:
                rel_msg_kernel<1><<<grid, block, 0, stream>>>(
                    node, atoms[r], W[r][0], W[r][1], W[r][2], W[r][3],
                    out + msg_base, natom);
                break;
            case 2:
                rel_msg_kernel<2><<<grid, block, 0, stream>>>(
                    node, atoms[r], W[r][0], W[r][1], W[r][2], W[r][3],
                    out + msg_base, natom);
                break;
            default:
                rel_msg_kernel<3><<<grid, block, 0, stream>>>(
                    node, atoms[r], W[r][0], W[r][1], W[r][2], W[r][3],
                    out + msg_base, natom);
                break;
        }
        msg_base += (size_t)flat * 64;        // flat rows of E=64 floats
    }

    // ---- index concat tail (int32 or int64 slot width, inferred from out_size)
    long long tot_idx = 0;
    for (int r = 0; r < 4; ++r) tot_idx += in_sizes[1 + r];
    long long tail_words = (long long)out_size - (long long)msg_base;

    if (tail_words >= 2 * tot_idx) {          // int64 output slots
        long long* dst = (long long*)(out + msg_base);
        long long off = 0;
        for (int r = 0; r < 4; ++r) {
            int n = in_sizes[1 + r];
            copy_idx64<<<(n + 255) / 256, 256, 0, stream>>>(atoms[r], n, dst + off);
            off += n;
        }
    } else if (tail_words >= tot_idx) {       // int32 output slots
        int* dst = (int*)(out + msg_base);
        long long off = 0;
        for (int r = 0; r < 4; ++r) {
            int n = in_sizes[1 + r];
            copy_idx32<<<(n + 255) / 256, 256, 0, stream>>>(atoms[r], n, dst + off);
            off += n;
        }
    }
}